// PathwayGraphEmbedding_89962384982526
// MI455X (gfx1250) — compile-verified
//
#include <hip/hip_runtime.h>
#include <hip/hip_bf16.h>
#include <stdint.h>

typedef __attribute__((ext_vector_type(2))) float v2f;
typedef __attribute__((ext_vector_type(8))) float v8f;

#define FDIM 128   // F == H == 128
#define NGENES 20000

// ---------------------------------------------------------------------------
// 1) degree: init to 1.0 (self loop), atomic count over dst, then rsqrt
// ---------------------------------------------------------------------------
__global__ void k_init_deg(float* __restrict__ deg, int n) {
    int i = blockIdx.x * blockDim.x + threadIdx.x;
    if (i < n) deg[i] = 1.0f;
}

__global__ void k_count_deg(const int* __restrict__ dst, float* __restrict__ deg, int e) {
    int i = blockIdx.x * blockDim.x + threadIdx.x;
    if (i < e) atomicAdd(&deg[dst[i]], 1.0f);
}

__global__ void k_rsqrt_inplace(float* __restrict__ d, int n) {
    int i = blockIdx.x * blockDim.x + threadIdx.x;
    if (i < n) d[i] = rsqrtf(d[i]);
}

// ---------------------------------------------------------------------------
// 2) gather: X[b,i,:] = gene_emb[b, gid[i], :]   (float4 per thread)
// ---------------------------------------------------------------------------
__global__ void k_gather(const float* __restrict__ gene_emb,
                         const int* __restrict__ gid,
                         float* __restrict__ X, int n) {
    long long t  = (long long)blockIdx.x * blockDim.x + threadIdx.x; // B*N*32 threads
    int  q   = (int)(t & 31);          // which float4 of the 128-float row
    long long bi = t >> 5;             // b*N + i
    int  i   = (int)(bi % n);
    int  b   = (int)(bi / n);
    int  g   = gid[i];
    const float4* srcp = (const float4*)(gene_emb + ((long long)b * NGENES + g) * FDIM);
    float4* dstp = (float4*)(X + bi * FDIM);
    dstp[q] = srcp[q];
}

// ---------------------------------------------------------------------------
// 3) zero a float buffer (float4 per thread)
// ---------------------------------------------------------------------------
__global__ void k_zero4(float* __restrict__ p) {
    long long t = (long long)blockIdx.x * blockDim.x + threadIdx.x;
    ((float4*)p)[t] = make_float4(0.f, 0.f, 0.f, 0.f);
}

// ---------------------------------------------------------------------------
// 4) GEMM via V_WMMA_F32_16X16X4_F32:  Hout = X(R,128) @ W(128,128)
//    block = 256 threads = 8 waves; wave w -> 16-col tile w; block -> 16 rows.
//    No divergence anywhere (WMMA needs EXEC all-ones). R % 16 == 0.
//    A frag (16x4 f32): lane l holds M=l&15, K = 4*ks + 2*(l>>4) + v  (v=vgpr)
//    B frag (4x16 f32): lane l holds N=l&15, same K indexing (mirror of A)
//    D (16x16 f32, 8 VGPRs): vgpr r, lane l -> N=l&15, M = r + 8*(l>>4)
// ---------------------------------------------------------------------------
__global__ void k_gemm_wmma(const float* __restrict__ X,
                            const float* __restrict__ W,
                            float* __restrict__ Hout) {
    const int lane  = threadIdx.x & 31;
    const int wave  = threadIdx.x >> 5;      // 0..7
    const int m     = lane & 15;             // row within tile / col within tile
    const int khalf = lane >> 4;             // 0 or 1
    const int col0  = wave * 16;
    const long long row0 = (long long)blockIdx.x * 16;

    const float* __restrict__ xrow = X + (row0 + m) * FDIM;

    v8f acc = {};
#pragma unroll 8
    for (int ks = 0; ks < 32; ++ks) {
        const int kb = ks * 4 + khalf * 2;
        v2f a = *(const v2f*)(xrow + kb);                 // A[m][kb], A[m][kb+1]
        v2f b;
        b.x = W[(long long)(kb    ) * FDIM + col0 + m];   // B[kb  ][col0+m]
        b.y = W[(long long)(kb + 1) * FDIM + col0 + m];   // B[kb+1][col0+m]
        acc = __builtin_amdgcn_wmma_f32_16x16x4_f32(
                  false, a, false, b, (short)0, acc, false, false);
    }
#pragma unroll
    for (int r = 0; r < 8; ++r) {
        const long long orow = row0 + r + 8 * khalf;      // C/D layout: M = r + 8*(l>>4)
        Hout[orow * FDIM + col0 + m] = acc[r];
    }
}

// ---------------------------------------------------------------------------
// 5) edge scatter: one wave per (edge-or-selfloop, batch); each lane: float4
//    agg[b,dst,:] += h[b,src,:] * dinv[src]*dinv[dst]
// ---------------------------------------------------------------------------
__global__ void k_scatter(const float* __restrict__ Hh,
                          const int* __restrict__ src,
                          const int* __restrict__ dst,
                          const float* __restrict__ dinv,
                          float* __restrict__ Agg,
                          int e, int n, int bcount) {
    long long t    = (long long)blockIdx.x * blockDim.x + threadIdx.x;
    long long pair = t >> 5;                 // (edge, batch) pair id
    int  lane = (int)(t & 31);
    long long total = (long long)(e + n) * bcount;
    if (pair >= total) return;

    int b = (int)(pair % bcount);
    long long ed = pair / bcount;
    int s, d;
    if (ed < e) { s = src[ed]; d = dst[ed]; }
    else        { s = d = (int)(ed - e); }   // self loop

    const float nrm = dinv[s] * dinv[d];
    const float4 hv = ((const float4*)(Hh + ((long long)b * n + s) * FDIM))[lane];
    float* ap = Agg + ((long long)b * n + d) * FDIM + lane * 4;
    atomicAdd(ap + 0, hv.x * nrm);
    atomicAdd(ap + 1, hv.y * nrm);
    atomicAdd(ap + 2, hv.z * nrm);
    atomicAdd(ap + 3, hv.w * nrm);
}

// ---------------------------------------------------------------------------
// 6) bias + relu (float4 per thread): X = relu(Agg + b1)
// ---------------------------------------------------------------------------
__global__ void k_bias_relu4(const float* __restrict__ Agg,
                             const float* __restrict__ bias,
                             float* __restrict__ Xout) {
    long long t = (long long)blockIdx.x * blockDim.x + threadIdx.x;
    int q = (int)(t & 31);                   // float4 index within a 128-row
    float4 v  = ((const float4*)Agg)[t];
    float4 bb = ((const float4*)bias)[q];
    v.x = fmaxf(v.x + bb.x, 0.f);
    v.y = fmaxf(v.y + bb.y, 0.f);
    v.z = fmaxf(v.z + bb.z, 0.f);
    v.w = fmaxf(v.w + bb.w, 0.f);
    ((float4*)Xout)[t] = v;
}

// ---------------------------------------------------------------------------
// 7) final mean: out[b,h] = b2[h] + (1/N) * sum_i Agg[b,i,h]
// ---------------------------------------------------------------------------
__global__ void k_init_out(const float* __restrict__ b2, float* __restrict__ out, int bcount) {
    int t = blockIdx.x * blockDim.x + threadIdx.x;      // bcount*128 threads
    if (t < bcount * FDIM) out[t] = b2[t & (FDIM - 1)];
}

__global__ void k_mean_partial(const float* __restrict__ Agg,
                               float* __restrict__ out,
                               int n, int rows_per_chunk) {
    const int h = threadIdx.x;               // 0..127
    const int b = blockIdx.y;
    const int r0 = blockIdx.x * rows_per_chunk;
    float s = 0.f;
    const float inv = 1.0f / (float)n;
    for (int r = r0; r < r0 + rows_per_chunk && r < n; ++r)
        s += Agg[((long long)b * n + r) * FDIM + h];
    atomicAdd(&out[b * FDIM + h], s * inv);
}

// ---------------------------------------------------------------------------
extern "C" void kernel_launch(void* const* d_in, const int* in_sizes, int n_in,
                              void* d_out, int out_size, void* d_ws, size_t ws_size,
                              hipStream_t stream) {
    const float* gene_emb = (const float*)d_in[0];   // (B, 20000, 128)
    const float* W1       = (const float*)d_in[1];   // (128,128)
    const float* b1       = (const float*)d_in[2];   // (128,)
    const float* W2       = (const float*)d_in[3];   // (128,128)
    const float* b2       = (const float*)d_in[4];   // (128,)
    const int*   gid      = (const int*)d_in[5];     // (N,)
    const int*   edges    = (const int*)d_in[6];     // (2,E) flat: src then dst

    const int N = in_sizes[5];
    const int E = in_sizes[6] / 2;
    const int B = in_sizes[0] / (NGENES * FDIM);
    const int* srcI = edges;
    const int* dstI = edges + E;

    float* out = (float*)d_out;

    // workspace layout
    const size_t big   = (size_t)B * N * FDIM * sizeof(float);   // ~205 MB each
    char* ws = (char*)d_ws;
    float* dinv = (float*)ws;                                    // N floats
    size_t o    = ((size_t)N * sizeof(float) + 255) & ~(size_t)255;
    float* X    = (float*)(ws + o);
    float* Hb   = (float*)(ws + o + big);
    float* Agg  = (float*)(ws + o + 2 * big);

    const long long R      = (long long)B * N;            // rows, multiple of 16
    const long long elems  = R * FDIM;                    // B*N*128
    const long long vec4   = elems / 4;                   // float4 count
    const long long pairs  = (long long)(E + N) * B;      // scatter pairs

    const int TB = 256;
    dim3 blk(TB);

    // --- degree / normalization ---
    k_init_deg<<<dim3((N + TB - 1) / TB), blk, 0, stream>>>(dinv, N);
    k_count_deg<<<dim3((E + TB - 1) / TB), blk, 0, stream>>>(dstI, dinv, E);
    k_rsqrt_inplace<<<dim3((N + TB - 1) / TB), blk, 0, stream>>>(dinv, N);

    // --- gather ---
    k_gather<<<dim3((unsigned)((R * 32 + TB - 1) / TB)), blk, 0, stream>>>(gene_emb, gid, X, N);

    // --- layer 1 ---
    k_gemm_wmma<<<dim3((unsigned)(R / 16)), blk, 0, stream>>>(X, W1, Hb);
    k_zero4<<<dim3((unsigned)((vec4 + TB - 1) / TB)), blk, 0, stream>>>(Agg);
    k_scatter<<<dim3((unsigned)((pairs * 32 + TB - 1) / TB)), blk, 0, stream>>>(
        Hb, srcI, dstI, dinv, Agg, E, N, B);
    k_bias_relu4<<<dim3((unsigned)((vec4 + TB - 1) / TB)), blk, 0, stream>>>(Agg, b1, X);

    // --- layer 2 ---
    k_gemm_wmma<<<dim3((unsigned)(R / 16)), blk, 0, stream>>>(X, W2, Hb);
    k_zero4<<<dim3((unsigned)((vec4 + TB - 1) / TB)), blk, 0, stream>>>(Agg);
    k_scatter<<<dim3((unsigned)((pairs * 32 + TB - 1) / TB)), blk, 0, stream>>>(
        Hb, srcI, dstI, dinv, Agg, E, N, B);

    // --- mean over N (+ b2) ---
    k_init_out<<<dim3((B * FDIM + TB - 1) / TB), blk, 0, stream>>>(b2, out, B);
    const int chunks = 250;
    const int rows_per_chunk = (N + chunks - 1) / chunks;
    k_mean_partial<<<dim3(chunks, B), dim3(FDIM), 0, stream>>>(Agg, out, N, rows_per_chunk);
}